// conjugate_gradient_loss_20572893347855
// MI455X (gfx1250) — compile-verified
//
#include <hip/hip_runtime.h>
#include <hip/hip_bf16.h>

// CG loss for K=128 systems of size N=1024 (fp32).
// Memory-bound: 12 batched matvecs x 512MB = 6.1 GB @ 23.3 TB/s ~ 0.26 ms floor.
// Matvec: async global->LDS staging (GLOBAL_LOAD_ASYNC_TO_LDS_B128, ASYNCcnt)
// feeding V_WMMA_F32_16X16X4_F32 out of a double-buffered, bank-padded LDS tile.

#define KSYS 128
#define NDIM 1024
#define KN (KSYS * NDIM)
#define PITCH 36   // 32 K-floats + 4 pad: 144B rows (16B aligned), conflict-free banks

typedef __attribute__((ext_vector_type(2))) float v2f;
typedef __attribute__((ext_vector_type(8))) float v8f;

// ---------------------------------------------------------------------------
// Y[sys] = M[sys] @ V[sys]
// One 128-thread workgroup (4 waves) per 64-row block; wave w owns rows
// [w*16, w*16+16). A tiles streamed via async b128 copies into LDS
// (each async instr: 8 consecutive lanes = one contiguous 128B line).
// WMMA A operand (16x4 f32): lane<16 -> (M=lane, K=0..1), lane>=16 -> K=2..3.
// B operand: vector chunk replicated across all 16 N columns, so every
// accumulator column equals A_tile @ v_chunk.
// ---------------------------------------------------------------------------
__global__ __launch_bounds__(128)
void matvec64_wmma_async(const float* __restrict__ M, const float* __restrict__ V,
                         float* __restrict__ Y) {
    __shared__ __align__(16) float sbuf[2][64 * PITCH];

    const int wg   = blockIdx.x;          // KSYS * 16 blocks
    const int sys  = wg >> 4;
    const int rb   = wg & 15;             // which 64-row block
    const int t    = threadIdx.x;         // 0..127
    const int wave = t >> 5;
    const int lane = t & 31;
    const int half = lane >> 4;           // 0: K pair {0,1}, 1: K pair {2,3}
    const int mrow = lane & 15;

    const float* __restrict__ Msys =
        M + (size_t)sys * NDIM * NDIM + (size_t)(rb * 64) * NDIM;
    const float* __restrict__ vp = V + sys * NDIM + 2 * half;

    // Stage s covers K = [s*32, s*32+32): 64 rows x 128B, 512 b128 chunks,
    // 4 async-load instructions per thread -> ASYNCcnt += 4 per wave.
    auto issue = [&](int s) {
        const int buf = s & 1;
        #pragma unroll
        for (int i = 0; i < 4; ++i) {
            const int li    = i * 128 + t;   // 0..511
            const int row   = li >> 3;       // 8 lanes per 128B row chunk
            const int col16 = li & 7;
            const float* g = Msys + (size_t)row * NDIM + s * 32 + col16 * 4;
            unsigned lds = (unsigned)(uintptr_t)&sbuf[buf][row * PITCH + col16 * 4];
            asm volatile("global_load_async_to_lds_b128 %0, %1, off"
                         :: "v"(lds), "v"(g) : "memory");
        }
    };

    v8f acc = {};
    issue(0);
    for (int s = 0; s < 32; ++s) {
        if (s + 1 < 32) {
            issue(s + 1);                                  // prefetch next tile
            asm volatile("s_wait_asynccnt 4" ::: "memory"); // stage s landed
        } else {
            asm volatile("s_wait_asynccnt 0" ::: "memory");
        }
        __syncthreads();   // stage-s tile visible to all 4 waves

        const float* arow = &sbuf[s & 1][(wave * 16 + mrow) * PITCH + 2 * half];
        const float* vk   = vp + s * 32;
        #pragma unroll
        for (int j = 0; j < 32; j += 4) {
            v2f a = *(const v2f*)(arow + j);   // ds_load_b64, conflict-free
            v2f b = *(const v2f*)(vk + j);     // 4KB vector, cache-resident
            // 8 args: (neg_a, A, neg_b, B, c_mod, C, reuse_a, reuse_b)
            acc = __builtin_amdgcn_wmma_f32_16x16x4_f32(
                false, a, false, b, (short)0, acc, false, false);
        }
        __syncthreads();   // all waves done reading buf[s&1] before overwrite
    }

    // C/D layout: VGPR r, lanes 0-15 hold M=r, lanes 16-31 hold M=8+r; all
    // columns identical -> lane 0 writes rows 0..7, lane 16 writes rows 8..15.
    if (mrow == 0) {
        float* yp = Y + sys * NDIM + rb * 64 + wave * 16 + 8 * half;
        #pragma unroll
        for (int r = 0; r < 8; ++r) yp[r] = acc[r];
    }
}

// ---------------------------------------------------------------------------
// out[sys] = sum_i X[sys,i]*Y[sys,i]   (one 256-thread block per system)
// ---------------------------------------------------------------------------
__global__ __launch_bounds__(256)
void dot_per_system(const float* __restrict__ X, const float* __restrict__ Y,
                    float* __restrict__ out) {
    __shared__ float s[256];
    const int sys = blockIdx.x, t = threadIdx.x;
    const float* x = X + sys * NDIM;
    const float* y = Y + sys * NDIM;
    float a = 0.f;
    #pragma unroll
    for (int i = t; i < NDIM; i += 256) a += x[i] * y[i];
    s[t] = a; __syncthreads();
    for (int off = 128; off > 0; off >>= 1) {
        if (t < off) s[t] += s[t + off];
        __syncthreads();
    }
    if (t == 0) out[sys] = s[0];
}

// R = b - AX ; rz_prev = 1 ; stop2 = max(rtol*||b||, atol)^2 ; done = 0
__global__ __launch_bounds__(256)
void cg_init(const float* __restrict__ b, const float* __restrict__ AX,
             const float* __restrict__ bb, float* __restrict__ R,
             float* __restrict__ rz_prev, float* __restrict__ stop2,
             int* __restrict__ done) {
    const int i = blockIdx.x * 256 + threadIdx.x;
    if (i < KN) R[i] = b[i] - AX[i];
    if (i < KSYS) {
        float st = fmaxf(1e-5f * sqrtf(bb[i]), 1e-5f);
        stop2[i] = st * st;
        rz_prev[i] = 1.0f;
    }
    if (i == 0) *done = 0;
}

// P = Z (first iter) or Z + (rz/safe(rz_prev)) * P   (frozen when done)
__global__ __launch_bounds__(256)
void update_p(const float* __restrict__ Z, float* __restrict__ P,
              const float* __restrict__ rz, const float* __restrict__ rz_prev,
              const int* __restrict__ done, int first) {
    if (*done) return;
    const int i = blockIdx.x * 256 + threadIdx.x;
    if (i >= KN) return;
    if (first) { P[i] = Z[i]; return; }
    const int sys = i >> 10;
    float d = rz_prev[sys]; d = (d == 0.f) ? 1e-8f : d;
    P[i] = Z[i] + (rz[sys] / d) * P[i];
}

// alpha = rz/safe(p.Ap) ; X += alpha P ; R -= alpha AP   (frozen when done)
__global__ __launch_bounds__(256)
void update_xr(float* __restrict__ X, float* __restrict__ R,
               const float* __restrict__ P, const float* __restrict__ AP,
               const float* __restrict__ rz, const float* __restrict__ pAp,
               const int* __restrict__ done) {
    if (*done) return;
    const int i = blockIdx.x * 256 + threadIdx.x;
    if (i >= KN) return;
    const int sys = i >> 10;
    float d = pAp[sys]; d = (d == 0.f) ? 1e-8f : d;
    const float alpha = rz[sys] / d;
    X[i] += alpha * P[i];
    R[i] -= alpha * AP[i];
}

// one block, K threads: rz_prev <- rz (if !done), then done |= all(||R||<=stop)
__global__ __launch_bounds__(KSYS)
void check_conv(const float* __restrict__ rr, const float* __restrict__ stop2,
                const float* __restrict__ rz, float* __restrict__ rz_prev,
                int* __restrict__ done) {
    __shared__ int allc;
    const int t = threadIdx.x;
    const int d0 = *done;
    if (t == 0) allc = 1;
    __syncthreads();
    if (!d0) rz_prev[t] = rz[t];
    if (!(rr[t] <= stop2[t])) atomicAnd(&allc, 0);
    __syncthreads();
    if (t == 0 && allc) *done = 1;
}

// loss = mean((X - 1)^2) over all K*N elements (single 1024-thread block)
__global__ __launch_bounds__(1024)
void loss_kernel(const float* __restrict__ X, float* __restrict__ out) {
    __shared__ float s[1024];
    const int t = threadIdx.x;
    float a = 0.f;
    for (int i = t; i < KN; i += 1024) {
        float d = X[i] - 1.0f;
        a += d * d;
    }
    s[t] = a; __syncthreads();
    for (int off = 512; off > 0; off >>= 1) {
        if (t < off) s[t] += s[t + off];
        __syncthreads();
    }
    if (t == 0) out[0] = s[0] / (float)KN;
}

extern "C" void kernel_launch(void* const* d_in, const int* in_sizes, int n_in,
                              void* d_out, int out_size, void* d_ws, size_t ws_size,
                              hipStream_t stream) {
    const float* A_inv = (const float*)d_in[0];
    const float* A     = (const float*)d_in[1];
    const float* b     = (const float*)d_in[2];

    float* ws = (float*)d_ws;                 // ~2.6 MB used
    float* X       = ws;
    float* R       = ws + 1 * KN;
    float* P       = ws + 2 * KN;
    float* Z       = ws + 3 * KN;
    float* AP      = ws + 4 * KN;             // also A@X scratch at init
    float* rz      = ws + 5 * KN;
    float* pAp     = rz + KSYS;
    float* rr      = pAp + KSYS;
    float* rz_prev = rr + KSYS;
    float* stop2   = rz_prev + KSYS;
    int*   done    = (int*)(stop2 + KSYS);

    const int mvGrid = KSYS * (NDIM / 64);    // one 4-wave WG per 64-row block
    const int vGrid  = (KN + 255) / 256;

    // X0 = A_inv @ b ; AX0 = A @ X0 ; R = b - AX0 ; stop from ||b||
    matvec64_wmma_async<<<mvGrid, 128, 0, stream>>>(A_inv, b, X);
    matvec64_wmma_async<<<mvGrid, 128, 0, stream>>>(A, X, AP);
    dot_per_system<<<KSYS, 256, 0, stream>>>(b, b, rr);   // rr <- ||b||^2 (temp)
    cg_init<<<vGrid, 256, 0, stream>>>(b, AP, rr, R, rz_prev, stop2, done);

    for (int it = 1; it <= 5; ++it) {
        matvec64_wmma_async<<<mvGrid, 128, 0, stream>>>(A_inv, R, Z);   // Z = M@R
        dot_per_system<<<KSYS, 256, 0, stream>>>(R, Z, rz);             // rz
        update_p<<<vGrid, 256, 0, stream>>>(Z, P, rz, rz_prev, done, it == 1);
        matvec64_wmma_async<<<mvGrid, 128, 0, stream>>>(A, P, AP);      // AP = A@P
        dot_per_system<<<KSYS, 256, 0, stream>>>(P, AP, pAp);           // p.Ap
        update_xr<<<vGrid, 256, 0, stream>>>(X, R, P, AP, rz, pAp, done);
        dot_per_system<<<KSYS, 256, 0, stream>>>(R, R, rr);             // ||R_new||^2
        check_conv<<<1, KSYS, 0, stream>>>(rr, stop2, rz, rz_prev, done);
    }

    loss_kernel<<<1, 1024, 0, stream>>>(X, (float*)d_out);
}